// TrilinearFusion_A_32959579029895
// MI455X (gfx1250) — compile-verified
//
#include <hip/hip_runtime.h>
#include <math.h>

typedef __attribute__((ext_vector_type(2))) float v2f;
typedef __attribute__((ext_vector_type(8))) float v8f;

#define D 64
#define DP1 65
#define BATCH 512
#define NOUT 256
#define F_TOT 274625UL   // 65^3
#define IJ 4225          // 65*65
#define SKIP 195

// ---------------- k0: zero the fp32 accumulator ----------------
__global__ void k0_zero(float* __restrict__ c, int n) {
    int t = blockIdx.x * 256 + threadIdx.x;
    if (t < n) c[t] = 0.0f;
}

// ---------------- k1a: z_l[b,o] = vec_x1 . zw_l[o] . vec3 + zb_l ----------------
// grid (16 btile, 3 branch, 2 o-half), block 256 = 32 batch x 8 i-groups
__global__ __launch_bounds__(256) void k1a_bilinear(
    const float* __restrict__ vec1, const float* __restrict__ vec2,
    const float* __restrict__ vec3,
    const float* __restrict__ zw1, const float* __restrict__ zb1,
    const float* __restrict__ zw2, const float* __restrict__ zb2,
    const float* __restrict__ zw3, const float* __restrict__ zb3,
    float* __restrict__ Z /* [3][512][64] */) {
    int btile = blockIdx.x, br = blockIdx.y, oh = blockIdx.z;
    const float* x1 = (br == 0) ? vec1 : (br == 1) ? vec2 : vec1;  // z3 uses vec1!
    const float* x2 = vec3;
    const float* zw = (br == 0) ? zw1 : (br == 1) ? zw2 : zw3;
    const float* zb = (br == 0) ? zb1 : (br == 1) ? zb2 : zb3;

    int tid = threadIdx.x;
    int bl = tid >> 3;      // 0..31
    int sub = tid & 7;      // i-group
    int b = btile * 32 + bl;

    __shared__ float s1[32 * DP1];
    __shared__ float s2[32 * DP1];
    for (int t = tid; t < 32 * D; t += 256) {
        int rb = t >> 6, c = t & 63;
        s1[rb * DP1 + c] = x1[(btile * 32 + rb) * D + c];
        s2[rb * DP1 + c] = x2[(btile * 32 + rb) * D + c];
    }
    __syncthreads();

    for (int o = oh * 32; o < oh * 32 + 32; ++o) {
        float acc = 0.0f;
        const float* zwo = zw + (size_t)o * (D * D);
        for (int ii = 0; ii < 8; ++ii) {
            int i = sub * 8 + ii;
            const float* zwrow = zwo + i * D;
            float t = 0.0f;
#pragma unroll 8
            for (int j = 0; j < D; ++j) t += zwrow[j] * s2[bl * DP1 + j];
            acc += s1[bl * DP1 + i] * t;
        }
        // reduce across the 8 i-group lanes (contiguous in the wave32)
        for (int m = 1; m < 8; m <<= 1) acc += __shfl_xor(acc, m, 32);
        if (sub == 0) Z[((size_t)br * BATCH + b) * D + o] = acc + zb[o];
    }
}

// ---------------- k1b: h, gate, output linear + relu, append 1 ----------------
// grid (128 batch-quads, 3 branches), block 256 = 4 batch x 64 out
__global__ __launch_bounds__(256) void k1b_gate(
    const float* __restrict__ vec1, const float* __restrict__ vec2,
    const float* __restrict__ vec3,
    const float* __restrict__ hw1, const float* __restrict__ hb1,
    const float* __restrict__ ow1, const float* __restrict__ ob1,
    const float* __restrict__ hw2, const float* __restrict__ hb2,
    const float* __restrict__ ow2, const float* __restrict__ ob2,
    const float* __restrict__ hw3, const float* __restrict__ hb3,
    const float* __restrict__ ow3, const float* __restrict__ ob3,
    const float* __restrict__ Z, float* __restrict__ Oall /* [3][512][65] */) {
    int bq = blockIdx.x, br = blockIdx.y;
    const float* hv = (br == 0) ? vec1 : (br == 1) ? vec2 : vec3;
    const float* hw = (br == 0) ? hw1 : (br == 1) ? hw2 : hw3;
    const float* hb = (br == 0) ? hb1 : (br == 1) ? hb2 : hb3;
    const float* ow = (br == 0) ? ow1 : (br == 1) ? ow2 : ow3;
    const float* ob = (br == 0) ? ob1 : (br == 1) ? ob2 : ob3;

    int tid = threadIdx.x;
    int bl = tid >> 6;  // 0..3
    int o = tid & 63;
    int b = bq * 4 + bl;

    __shared__ float sv[4 * DP1];
    __shared__ float g[4 * DP1];
    sv[bl * DP1 + o] = hv[(size_t)b * D + o];
    __syncthreads();

    float h = hb[o];
#pragma unroll 8
    for (int i = 0; i < D; ++i) h += hw[o * D + i] * sv[bl * DP1 + i];
    h = fmaxf(h, 0.0f);
    float z = Z[((size_t)br * BATCH + b) * D + o];
    float sg = 1.0f / (1.0f + expf(-z));
    g[bl * DP1 + o] = sg * h;
    __syncthreads();

    float acc = ob[o];
#pragma unroll 8
    for (int i = 0; i < D; ++i) acc += ow[o * D + i] * g[bl * DP1 + i];
    acc = fmaxf(acc, 0.0f);
    Oall[((size_t)br * BATCH + b) * DP1 + o] = acc;
    if (o == 0) Oall[((size_t)br * BATCH + b) * DP1 + D] = 1.0f;
}

// ---------------- k2: enc1 GEMM with on-the-fly trilinear A, fp32 WMMA ----------------
// grid (65 i-slices, 8 batch tiles of 64), block 256 = 8 waves.
// Wave w owns output columns [32w, 32w+32) as 2 N-tiles; 4 M-tiles of 16 batches.
static __device__ inline v8f wmma_f32(v2f a, v2f b, v8f c) {
    return __builtin_amdgcn_wmma_f32_16x16x4_f32(false, a, false, b, (short)0, c,
                                                 false, false);
}

__global__ __launch_bounds__(256) void k2_enc1(
    const float* __restrict__ W /* [256][F] */,
    const float* __restrict__ Oall /* [3][512][65] */,
    float* __restrict__ Cacc /* [512][256] */) {
    const int i = blockIdx.x;        // 0..64 (o1 index)
    const int btile = blockIdx.y;    // 0..7  (64 batches)
    const int tid = threadIdx.x;
    const int wave = tid >> 5;
    const int lane = tid & 31;
    const int lr = lane & 15;
    const int lh = lane >> 4;        // K half: kb = 2*lh
    const int kb = 2 * lh;

    __shared__ float o1c[64];
    __shared__ float o2t[64 * DP1];
    __shared__ float o3t[64 * DP1];

    const float* o1g = Oall + (size_t)0 * BATCH * DP1 + (size_t)(btile * 64) * DP1;
    const float* o2g = Oall + (size_t)1 * BATCH * DP1 + (size_t)(btile * 64) * DP1;
    const float* o3g = Oall + (size_t)2 * BATCH * DP1 + (size_t)(btile * 64) * DP1;
    for (int t = tid; t < 64 * DP1; t += 256) {
        o2t[t] = o2g[t];
        o3t[t] = o3g[t];
    }
    if (tid < 64) o1c[tid] = o1g[tid * DP1 + i];
    __syncthreads();

    const int n0 = wave * 32;
    const float* wrow0 = W + (size_t)(n0 + lr) * F_TOT + (size_t)i * IJ;
    const float* wrow1 = W + (size_t)(n0 + 16 + lr) * F_TOT + (size_t)i * IJ;

    v8f acc[4][2] = {};

    // j = 0..63 in 16 groups of 4; each group folds its four k=64 remainders
    // (o3[b,64]==1) into one extra K=4 WMMA.
    for (int jg = 0; jg < 16; ++jg) {
        float aremx[4] = {0.f, 0.f, 0.f, 0.f};
        float aremy[4] = {0.f, 0.f, 0.f, 0.f};
#pragma unroll
        for (int jj = 0; jj < 4; ++jj) {
            const int j = jg * 4 + jj;
            const int joff = j * DP1;
            float p12[4];
#pragma unroll
            for (int m = 0; m < 4; ++m) {
                int bl = m * 16 + lr;
                p12[m] = o1c[bl] * o2t[bl * DP1 + j];
                aremx[m] = (jj == kb) ? p12[m] : aremx[m];
                aremy[m] = (jj == kb + 1) ? p12[m] : aremy[m];
            }
#pragma unroll 4
            for (int k0 = 0; k0 < 64; k0 += 4) {
                const int off = joff + k0 + kb;
                v2f bf0, bf1;
                bf0.x = wrow0[off]; bf0.y = wrow0[off + 1];
                bf1.x = wrow1[off]; bf1.y = wrow1[off + 1];
#pragma unroll
                for (int m = 0; m < 4; ++m) {
                    int bl = m * 16 + lr;
                    v2f a;
                    a.x = p12[m] * o3t[bl * DP1 + k0 + kb];
                    a.y = p12[m] * o3t[bl * DP1 + k0 + kb + 1];
                    acc[m][0] = wmma_f32(a, bf0, acc[m][0]);
                    acc[m][1] = wmma_f32(a, bf1, acc[m][1]);
                }
            }
        }
        // remainder WMMA: K slots = the four j's k=64 terms
        const int jA = jg * 4 + kb, jB = jA + 1;
        v2f br0, br1;
        br0.x = wrow0[jA * DP1 + D]; br0.y = wrow0[jB * DP1 + D];
        br1.x = wrow1[jA * DP1 + D]; br1.y = wrow1[jB * DP1 + D];
#pragma unroll
        for (int m = 0; m < 4; ++m) {
            v2f a; a.x = aremx[m]; a.y = aremy[m];
            acc[m][0] = wmma_f32(a, br0, acc[m][0]);
            acc[m][1] = wmma_f32(a, br1, acc[m][1]);
        }
    }

    // tail j = 64 (o2 ones column): full k part + single-slot remainder
    {
        const int joff = D * DP1;
        float p12[4];
#pragma unroll
        for (int m = 0; m < 4; ++m) {
            int bl = m * 16 + lr;
            p12[m] = o1c[bl] * o2t[bl * DP1 + D];
        }
#pragma unroll 4
        for (int k0 = 0; k0 < 64; k0 += 4) {
            const int off = joff + k0 + kb;
            v2f bf0, bf1;
            bf0.x = wrow0[off]; bf0.y = wrow0[off + 1];
            bf1.x = wrow1[off]; bf1.y = wrow1[off + 1];
#pragma unroll
            for (int m = 0; m < 4; ++m) {
                int bl = m * 16 + lr;
                v2f a;
                a.x = p12[m] * o3t[bl * DP1 + k0 + kb];
                a.y = p12[m] * o3t[bl * DP1 + k0 + kb + 1];
                acc[m][0] = wmma_f32(a, bf0, acc[m][0]);
                acc[m][1] = wmma_f32(a, bf1, acc[m][1]);
            }
        }
        // k = 64 term: only K slot 0 is live (A zero elsewhere -> B value irrelevant)
        v2f br0, br1;
        br0.x = wrow0[joff + D]; br0.y = br0.x;
        br1.x = wrow1[joff + D]; br1.y = br1.x;
#pragma unroll
        for (int m = 0; m < 4; ++m) {
            v2f a;
            a.x = (lh == 0) ? p12[m] : 0.0f;
            a.y = 0.0f;
            acc[m][0] = wmma_f32(a, br0, acc[m][0]);
            acc[m][1] = wmma_f32(a, br1, acc[m][1]);
        }
    }

    // split-K accumulation: hardware fp32 atomics into Cacc
    const int nA = n0 + lr, nB = n0 + 16 + lr;
#pragma unroll
    for (int m = 0; m < 4; ++m) {
#pragma unroll
        for (int r = 0; r < 8; ++r) {
            int b = btile * 64 + m * 16 + r + 8 * lh;
            unsafeAtomicAdd(&Cacc[(size_t)b * NOUT + nA], acc[m][0][r]);
            unsafeAtomicAdd(&Cacc[(size_t)b * NOUT + nB], acc[m][1][r]);
        }
    }
}

// ---------------- k3: bias+relu, skip concat, enc2 linear + relu ----------------
__global__ __launch_bounds__(256) void k3_enc2(
    const float* __restrict__ Cacc, const float* __restrict__ e1b,
    const float* __restrict__ Oall, const float* __restrict__ e2w,
    const float* __restrict__ e2b, float* __restrict__ out) {
    int b = blockIdx.x;
    int o = threadIdx.x;  // 0..255
    __shared__ float s[NOUT + SKIP];
    s[o] = fmaxf(Cacc[(size_t)b * NOUT + o] + e1b[o], 0.0f);
    if (o < SKIP) {
        int br = o / DP1, c = o % DP1;
        s[NOUT + o] = Oall[((size_t)br * BATCH + b) * DP1 + c];
    }
    __syncthreads();
    float acc = e2b[o];
    const float* wr = e2w + (size_t)o * (NOUT + SKIP);
#pragma unroll 4
    for (int k = 0; k < NOUT + SKIP; ++k) acc += wr[k] * s[k];
    out[(size_t)b * NOUT + o] = fmaxf(acc, 0.0f);
}

// ---------------- launch ----------------
extern "C" void kernel_launch(void* const* d_in, const int* in_sizes, int n_in,
                              void* d_out, int out_size, void* d_ws, size_t ws_size,
                              hipStream_t stream) {
    const float* vec1 = (const float*)d_in[0];
    const float* vec2 = (const float*)d_in[1];
    const float* vec3 = (const float*)d_in[2];
    const float* h1_w = (const float*)d_in[3];  const float* h1_b = (const float*)d_in[4];
    const float* z1_w = (const float*)d_in[5];  const float* z1_b = (const float*)d_in[6];
    const float* o1_w = (const float*)d_in[7];  const float* o1_b = (const float*)d_in[8];
    const float* h2_w = (const float*)d_in[9];  const float* h2_b = (const float*)d_in[10];
    const float* z2_w = (const float*)d_in[11]; const float* z2_b = (const float*)d_in[12];
    const float* o2_w = (const float*)d_in[13]; const float* o2_b = (const float*)d_in[14];
    const float* h3_w = (const float*)d_in[15]; const float* h3_b = (const float*)d_in[16];
    const float* z3_w = (const float*)d_in[17]; const float* z3_b = (const float*)d_in[18];
    const float* o3_w = (const float*)d_in[19]; const float* o3_b = (const float*)d_in[20];
    const float* enc1_w = (const float*)d_in[21]; const float* enc1_b = (const float*)d_in[22];
    const float* enc2_w = (const float*)d_in[23]; const float* enc2_b = (const float*)d_in[24];

    float* ws = (float*)d_ws;
    float* Z = ws;                           // 3*512*64  = 98304
    float* Oall = Z + 3 * BATCH * D;         // 3*512*65  = 99840
    float* Cacc = Oall + 3 * BATCH * DP1;    // 512*256   = 131072

    const int nC = BATCH * NOUT;
    k0_zero<<<(nC + 255) / 256, 256, 0, stream>>>(Cacc, nC);

    dim3 g1a(16, 3, 2);
    k1a_bilinear<<<g1a, 256, 0, stream>>>(vec1, vec2, vec3, z1_w, z1_b, z2_w, z2_b,
                                          z3_w, z3_b, Z);
    dim3 g1b(BATCH / 4, 3);
    k1b_gate<<<g1b, 256, 0, stream>>>(vec1, vec2, vec3, h1_w, h1_b, o1_w, o1_b, h2_w,
                                      h2_b, o2_w, o2_b, h3_w, h3_b, o3_w, o3_b, Z, Oall);

    dim3 g2(DP1, BATCH / 64);
    k2_enc1<<<g2, 256, 0, stream>>>(enc1_w, Oall, Cacc);

    k3_enc2<<<BATCH, NOUT, 0, stream>>>(Cacc, enc1_b, Oall, enc2_w, enc2_b,
                                        (float*)d_out);
}